// FusedRestoreNormBlock_14061722927565
// MI455X (gfx1250) — compile-verified
//
#include <hip/hip_runtime.h>

// MI455X (gfx1250): wave32, WMMA. Fused W@msg + residual + LayerNorm(H) + affine.
// Exact f32 math via V_WMMA_F32_16X16X4_F32 (memory-bound problem: 604MB moved,
// only ~17 GFLOP, so f32 matrix path is the right precision/perf tradeoff).

typedef __attribute__((ext_vector_type(2))) float v2f;
typedef __attribute__((ext_vector_type(8))) float v8f;

#define H_DIM  512
#define HRED   128
#define PIX    8192      // T*R*Co = 32*16*16 pixels per batch
#define TILE_N 16
#define WAVES  8
#define HT_PW  4         // 32 H-tiles of 16 rows / 8 waves

__launch_bounds__(256, 2)
__global__ void fused_restore_ln_kernel(const float* __restrict__ msg,
                                        const float* __restrict__ res,
                                        const float* __restrict__ wr,
                                        const float* __restrict__ gamma,
                                        const float* __restrict__ beta,
                                        float* __restrict__ out) {
    __shared__ float sMsg[HRED * TILE_N];   // [d][n], 8 KB, conflict-free for B frags
    __shared__ float sGamma[H_DIM];
    __shared__ float sBeta[H_DIM];
    __shared__ float sS1[WAVES][TILE_N];
    __shared__ float sS2[WAVES][TILE_N];

    const int tid = threadIdx.x;
    const int b   = blockIdx.y;
    const int p0  = blockIdx.x * TILE_N;

    const float* msgB = msg + (size_t)b * HRED  * PIX + p0;
    const float* resB = res + (size_t)b * H_DIM * PIX + p0;
    float*       outB = out + (size_t)b * H_DIM * PIX + p0;

    // ---- Stage msg tile (128x16 f32) as float4s + gamma/beta into LDS ----
    for (int q = tid; q < (HRED * TILE_N) / 4; q += 256) {
        const int d  = q >> 2;
        const int c4 = (q & 3) * 4;
        const float4 v = *(const float4*)(msgB + (size_t)d * PIX + c4);
        *(float4*)(&sMsg[d * TILE_N + c4]) = v;
    }
    for (int h = tid; h < H_DIM; h += 256) {
        sGamma[h] = gamma[h];
        sBeta[h]  = beta[h];
    }
    __syncthreads();

    // ---- GEMM: acc[i] = W[h0+i*16 .. +16, :] @ M[:, p0..p0+16] ----
    const int lane = tid & 31;
    const int wv   = tid >> 5;
    const int col  = lane & 15;             // N index (and A-matrix M index)
    const int kb   = (lane >> 4) << 1;      // K sub-offset: 0 (lanes 0-15) / 2 (16-31)
    const int h0   = wv * (HT_PW * 16);     // 64 rows per wave

    v8f acc[HT_PW] = {};

    // A row base for this lane: row = h0 + (lane%16), +i*16 rows per h-tile
    const float* wRow = wr + (size_t)(h0 + col) * HRED + kb;

    #pragma unroll
    for (int k0 = 0; k0 < HRED; k0 += 4) {
        v2f bf;                                   // B frag: rows k0+kb, k0+kb+1, col n
        bf.x = sMsg[(k0 + kb)     * TILE_N + col];
        bf.y = sMsg[(k0 + kb + 1) * TILE_N + col];
        #pragma unroll
        for (int i = 0; i < HT_PW; ++i) {
            const v2f af = *(const v2f*)(wRow + (size_t)i * 16 * HRED + k0);  // b64, 8B-aligned
            acc[i] = __builtin_amdgcn_wmma_f32_16x16x4_f32(
                         false, af, false, bf, (short)0, acc[i], false, false);
        }
    }

    // ---- Residual add + per-lane LN partial sums ----
    const int hHalf = (lane >> 4) * 8;      // C/D layout: lanes 16-31 hold rows +8
    float s1 = 0.0f, s2 = 0.0f;
    #pragma unroll
    for (int i = 0; i < HT_PW; ++i) {
        #pragma unroll
        for (int j = 0; j < 8; ++j) {
            const int    h   = h0 + i * 16 + j + hHalf;
            const size_t off = (size_t)h * PIX + col;
            const float  y   = acc[i][j] + resB[off];   // coalesced 64B segments
            acc[i][j] = y;
            s1 += y;
            s2 += y * y;
        }
    }

    // fold the two lane-halves of each column, then cross-wave via LDS
    s1 += __shfl_xor(s1, 16, 32);
    s2 += __shfl_xor(s2, 16, 32);
    if (lane < 16) {
        sS1[wv][lane] = s1;
        sS2[wv][lane] = s2;
    }
    __syncthreads();

    float t1 = 0.0f, t2 = 0.0f;
    #pragma unroll
    for (int w2 = 0; w2 < WAVES; ++w2) {
        t1 += sS1[w2][col];
        t2 += sS2[w2][col];
    }
    const float mean = t1 * (1.0f / H_DIM);
    const float var  = t2 * (1.0f / H_DIM) - mean * mean;
    const float rstd = rsqrtf(var + 1e-5f);

    // ---- Normalize + affine + coalesced store ----
    #pragma unroll
    for (int i = 0; i < HT_PW; ++i) {
        #pragma unroll
        for (int j = 0; j < 8; ++j) {
            const int    h   = h0 + i * 16 + j + hHalf;
            const size_t off = (size_t)h * PIX + col;
            outB[off] = (acc[i][j] - mean) * rstd * sGamma[h] + sBeta[h];
        }
    }
}

extern "C" void kernel_launch(void* const* d_in, const int* in_sizes, int n_in,
                              void* d_out, int out_size, void* d_ws, size_t ws_size,
                              hipStream_t stream) {
    const float* msg   = (const float*)d_in[0];  // (16,128,32,16,16)
    const float* res   = (const float*)d_in[1];  // (16,512,32,16,16)
    const float* wr    = (const float*)d_in[2];  // (512,128)
    const float* gamma = (const float*)d_in[3];  // (512,)
    const float* beta  = (const float*)d_in[4];  // (512,)
    float* out = (float*)d_out;                  // (16,512,32,16,16)

    dim3 grid(PIX / TILE_N, 16, 1);  // 512 pixel-tiles x 16 batches
    dim3 block(256, 1, 1);           // 8 waves
    fused_restore_ln_kernel<<<grid, block, 0, stream>>>(msg, res, wr, gamma, beta, out);
}